// HIPPOmod_8967891714399
// MI455X (gfx1250) — compile-verified
//
#include <hip/hip_runtime.h>
#include <hip/hip_bf16.h>

typedef float v2f __attribute__((ext_vector_type(2)));
typedef float v8f __attribute__((ext_vector_type(8)));

#define NC    64      // state dim
#define TLEN  1000
#define MDIM  128     // input dim
#define PITCH 65      // float pitch for At (one-time use)
#define P2H   80      // float2 pitch for hT2: 2*80 % 64 == 32 -> GEMM b64 loads conflict-free
#define P2X   68      // float2 pitch for xT2: 8*68 % 64 == 32 -> epilogue b64 stores conflict-free

// One persistent workgroup per (batch, half-of-M). 64 threads = 2 waves (wave32).
// h tile (64 rows of M x 64 state) lives in LDS (packed float2 along n) for the whole T loop.
__global__ __launch_bounds__(64) void hippo_legt_scan(
    const float* __restrict__ times,   // [T]
    const float* __restrict__ Y,       // [B, T, M]
    const float* __restrict__ mask,    // [B, T]
    const float* __restrict__ A,       // [N, N]
    const float* __restrict__ Bvec,    // [N]
    float* __restrict__ out)           // [B, M, N]
{
    __shared__ float At[NC * PITCH];     // At[n][k] = A[k][n]
    __shared__ v2f   hT2[32 * P2H];      // hT2[n2][m] = {h[m][2n2], h[m][2n2+1]}
    __shared__ v2f   xT2[32 * P2X];      // xT2[k2][m] = {x[m][2k2], x[m][2k2+1]}
    __shared__ float u_lds[NC];
    __shared__ float bv_lds[NC];
    __shared__ float r_lds[NC];          // r_j = sqrt(-A[j][j]); strict-lower A[k][j] = -r_k r_j
    __shared__ float dinv_lds[NC];       // 1 / (1 - 0.5*dt*A[j][j])

    const int tid  = threadIdx.x;        // 0..63
    const int lane = tid & 31;
    const int wave = tid >> 5;           // 0 or 1
    const int l16  = lane & 15;
    const int hi   = lane >> 4;          // 0: lanes 0-15, 1: lanes 16-31

    const int b  = blockIdx.x >> 1;
    const int m0 = (blockIdx.x & 1) * 64;

    // ---- init: transpose A into LDS, zero h, small vectors ----
    for (int i = tid; i < NC * NC; i += 64) {
        int k = i >> 6, n = i & 63;
        At[n * PITCH + k] = A[i];
    }
    for (int i = tid; i < 32 * P2H; i += 64) hT2[i] = (v2f)(0.0f);
    {
        bv_lds[tid] = Bvec[tid];
        float d = A[tid * NC + tid];
        r_lds[tid] = sqrtf(fmaxf(-d, 0.0f));
    }
    __syncthreads();

    // ---- preload constant WMMA A-fragments (A matrix rows = this wave's k-stripes) ----
    // 16x4 fp32 A-frag layout: lane M = l16, VGPR0 holds K = 2*hi, VGPR1 holds K = 2*hi+1.
    v2f aA[2][16];
#pragma unroll
    for (int s = 0; s < 2; ++s) {
        int krow = (wave * 2 + s) * 16 + l16;
#pragma unroll
        for (int ns = 0; ns < 16; ++ns) {
            int nb = ns * 4 + 2 * hi;
            aA[s][ns].x = At[nb * PITCH + krow];        // A[krow][nb]
            aA[s][ns].y = At[(nb + 1) * PITCH + krow];  // A[krow][nb+1]
        }
    }

    float prev_t = 0.0f;
    const size_t yb = (size_t)b * TLEN * MDIM + (size_t)m0;

    for (int t = 0; t < TLEN; ++t) {
        if (t + 8 < TLEN)
            __builtin_prefetch(&Y[yb + (size_t)(t + 8) * MDIM + tid], 0, 1);

        float mv = mask[b * TLEN + t];        // uniform across block
        if (mv != 0.0f) {
            float tt  = times[t];
            float dt  = (tt - prev_t) / tt;
            float hdt = 0.5f * dt;

            u_lds[tid] = dt * Y[yb + (size_t)t * MDIM + tid];
            {
                float ad = At[tid * PITCH + tid];          // A[j][j]
                dinv_lds[tid] = 1.0f / (1.0f - hdt * ad);
            }
            __syncthreads();

            // ---- x^T = h^T + hdt*(A h^T) + Bvec (x) (dt*u)   via v_wmma_f32_16x16x4_f32
            v8f acc[2][4];
#pragma unroll
            for (int s = 0; s < 2; ++s)
#pragma unroll
                for (int mt = 0; mt < 4; ++mt)
                    acc[s][mt] = (v8f)(0.0f);

#pragma unroll
            for (int ns = 0; ns < 16; ++ns) {
                // B-frag: one ds_load_b64 each; row n2 = 2*ns + hi holds {h[mc][nb], h[mc][nb+1]}
                v2f bf[4];
#pragma unroll
                for (int mt = 0; mt < 4; ++mt)
                    bf[mt] = hT2[(ns * 2 + hi) * P2H + mt * 16 + l16];
#pragma unroll
                for (int s = 0; s < 2; ++s)
#pragma unroll
                    for (int mt = 0; mt < 4; ++mt)
                        acc[s][mt] = __builtin_amdgcn_wmma_f32_16x16x4_f32(
                            false, aA[s][ns], false, bf[mt], (short)0, acc[s][mt], false, false);
            }

            // epilogue: D[kr][mc] = sum_n A[krow][n] h[mc][n]; combine and store x^T (paired)
#pragma unroll
            for (int s = 0; s < 2; ++s) {
                int n2b = (wave * 2 + s) * 8 + 4 * hi;   // (stripe*16 + 8*hi) / 2
#pragma unroll
                for (int mt = 0; mt < 4; ++mt) {
                    int mc = mt * 16 + l16;
                    float um = u_lds[mc];
#pragma unroll
                    for (int r2 = 0; r2 < 4; ++r2) {
                        int n2   = n2b + r2;
                        int krow = n2 * 2;
                        v2f hp = hT2[n2 * P2H + mc];
                        v2f xv;
                        xv.x = fmaf(hdt, acc[s][mt][2 * r2],     hp.x);
                        xv.x = fmaf(um,  bv_lds[krow],           xv.x);
                        xv.y = fmaf(hdt, acc[s][mt][2 * r2 + 1], hp.y);
                        xv.y = fmaf(um,  bv_lds[krow + 1],       xv.y);
                        xT2[n2 * P2X + mc] = xv;
                    }
                }
            }
            __syncthreads();

            // ---- forward substitution, O(N) per row via rank-1 lower structure:
            // L[k][j] = -hdt*A[k][j] = +hdt*r_k*r_j (k>j);  L[j][j] = 1 - hdt*A[j][j]
            // y_j = (x_j - hdt*r_j*sum_{j'<j} r_j' y_j') * dinv_j
            {
                const int m = tid;
                float xr[NC];
#pragma unroll
                for (int k2 = 0; k2 < 32; ++k2) {
                    v2f p = xT2[k2 * P2X + m];
                    xr[2 * k2]     = p.x;
                    xr[2 * k2 + 1] = p.y;
                }
                float sacc = 0.0f;
#pragma unroll
                for (int j = 0; j < NC; ++j) {
                    float rj  = r_lds[j];
                    float num = fmaf(-hdt * rj, sacc, xr[j]);
                    float yj  = num * dinv_lds[j];
                    xr[j] = yj;
                    sacc = fmaf(rj, yj, sacc);
                }
#pragma unroll
                for (int k2 = 0; k2 < 32; ++k2) {
                    v2f p;
                    p.x = xr[2 * k2];
                    p.y = xr[2 * k2 + 1];
                    hT2[k2 * P2H + m] = p;
                }
            }
            prev_t = tt;
            __syncthreads();
        }
    }

    // ---- write final h: out[b][m0+m][n] = h[m][n] (coalesced in n) ----
    __syncthreads();
    const float* hTf = (const float*)hT2;
    for (int i = tid; i < NC * NC; i += 64) {
        int m = i >> 6, n = i & 63;
        out[((size_t)(b * MDIM + m0 + m)) * NC + n] = hTf[((n >> 1) * P2H + m) * 2 + (n & 1)];
    }
}

extern "C" void kernel_launch(void* const* d_in, const int* in_sizes, int n_in,
                              void* d_out, int out_size, void* d_ws, size_t ws_size,
                              hipStream_t stream) {
    const float* times = (const float*)d_in[0];
    const float* Y     = (const float*)d_in[1];
    const float* mask  = (const float*)d_in[2];
    const float* A     = (const float*)d_in[3];
    const float* Bvec  = (const float*)d_in[4];
    float* out = (float*)d_out;

    const int B = in_sizes[2] / TLEN;       // mask is [B, T]
    dim3 grid(B * 2);                       // (batch, half-of-M) per workgroup
    dim3 block(64);                         // 2 waves (wave32)
    hippo_legt_scan<<<grid, block, 0, stream>>>(times, Y, mask, A, Bvec, out);
}